// SheafDiffuser_77077483094917
// MI455X (gfx1250) — compile-verified
//
#include <hip/hip_runtime.h>
#include <hip/hip_bf16.h>

// CDNA5 / gfx1250 wave32
typedef __attribute__((ext_vector_type(2))) float v2f;
typedef __attribute__((ext_vector_type(8))) float v8f;

#define N_NODES 100000
#define N_EDGES 1000000
#define HIDDEN  64
#define IN_DIM  4

// ---------------------------------------------------------------------------
// Phase 1: h = x @ W1 + b1 via V_WMMA_F32_16X16X4_F32 (exact shape match:
// M=16 nodes per wave, K=4=IN_DIM, N=16, 4 N-tiles for HIDDEN=64).
// Writes h into both `h` (read-only copy for gathers) and `acc` (scatter
// accumulator seed), so the final projection reads only `acc`.
// A-matrix 16x4 f32 layout (ISA 7.12.2): lane[0:15] rows M=0..15 hold K={0,1}
// in VGPR{0,1}; lanes[16:31] hold K={2,3}.  B 4x16: lane%16 = N, VGPR0/1 hold
// K = 2*(lane/16) + {0,1}.  C/D 16x16: lane%16 = N, VGPR r -> M = r + 8*(lane/16).
// ---------------------------------------------------------------------------
__global__ void __launch_bounds__(256)
sheaf_h_wmma(const float* __restrict__ x,
             const float* __restrict__ W1,
             const float* __restrict__ b1,
             float* __restrict__ h,
             float* __restrict__ acc,
             int n_nodes)
{
    const int wave = (blockIdx.x * blockDim.x + threadIdx.x) >> 5; // one 16-row tile per wave
    const int lane = threadIdx.x & 31;
    const int row0 = wave * 16;
    if (row0 >= n_nodes) return;            // wave-uniform: EXEC stays all-1s for WMMA

    const int m     = lane & 15;            // row-in-tile (A) / column N (B, C/D)
    const int khalf = lane >> 4;            // 0: K={0,1}, 1: K={2,3}
    const int row   = row0 + m;

    v2f a;
    a.x = x[row * IN_DIM + 2 * khalf + 0];
    a.y = x[row * IN_DIM + 2 * khalf + 1];

    #pragma unroll
    for (int nt = 0; nt < HIDDEN / 16; ++nt) {
        const int col = nt * 16 + m;        // this lane's N within full HIDDEN
        v2f b;
        b.x = W1[(2 * khalf + 0) * HIDDEN + col];
        b.y = W1[(2 * khalf + 1) * HIDDEN + col];
        v8f c = {};
        // 8 args: (neg_a, A, neg_b, B, c_mod, C, reuse_a, reuse_b)
        c = __builtin_amdgcn_wmma_f32_16x16x4_f32(
                false, a, false, b, (short)0, c, false, false);
        const float bias = b1[col];
        #pragma unroll
        for (int r = 0; r < 8; ++r) {
            const int   mr  = r + 8 * khalf;          // row of this D element
            const float val = c[r] + bias;
            const int   idx = (row0 + mr) * HIDDEN + col;
            h[idx]   = val;
            acc[idx] = val;
        }
    }
}

// ---------------------------------------------------------------------------
// Phase 2 (dominant cost, L2-resident): one wave per edge.  Lane L owns
// channels {2L, 2L+1} of both messages (float2 loads).  Lane 0 applies the
// 2x2 rotation to channels 0/1.  4 f32 atomic adds per lane into `acc`.
// ---------------------------------------------------------------------------
__global__ void __launch_bounds__(256)
sheaf_edge_diffuse(const float* __restrict__ h,
                   const int*   __restrict__ edges,
                   const float* __restrict__ phases,
                   float*       __restrict__ acc,
                   int n_edges)
{
    const int  gid  = blockIdx.x * blockDim.x + threadIdx.x;
    const int  e    = gid >> 5;
    const int  lane = gid & 31;
    if (e >= n_edges) return;

    const int u = edges[2 * e + 0];
    const int v = edges[2 * e + 1];
    float s, c;
    __sincosf(phases[e], &s, &c);

    const float2 au = ((const float2*)(h + (size_t)u * HIDDEN))[lane]; // hu ch {2L,2L+1}
    const float2 av = ((const float2*)(h + (size_t)v * HIDDEN))[lane]; // hv ch {2L,2L+1}

    float mv0 = au.x, mv1 = au.y;   // msg_v = R(+phase) on hu[0:2], identity elsewhere
    float mu0 = av.x, mu1 = av.y;   // msg_u = R(-phase) on hv[0:2], identity elsewhere
    if (lane == 0) {
        mv0 =  c * au.x - s * au.y;
        mv1 =  s * au.x + c * au.y;
        mu0 =  c * av.x + s * av.y;
        mu1 = -s * av.x + c * av.y;
    }

    float* av_ptr = acc + (size_t)v * HIDDEN + 2 * lane;
    float* au_ptr = acc + (size_t)u * HIDDEN + 2 * lane;
    atomicAdd(av_ptr + 0, mv0);
    atomicAdd(av_ptr + 1, mv1);
    atomicAdd(au_ptr + 0, mu0);
    atomicAdd(au_ptr + 1, mu1);
}

// ---------------------------------------------------------------------------
// Phase 3: out[node] = dot(acc[node,:], Wout) + bout.  One wave per node,
// 2 channels per lane, wave32 xor-shuffle reduction.
// ---------------------------------------------------------------------------
__global__ void __launch_bounds__(256)
sheaf_project(const float* __restrict__ acc,
              const float* __restrict__ Wout,
              const float* __restrict__ bout,
              float*       __restrict__ out,
              int n_nodes)
{
    const int gid  = blockIdx.x * blockDim.x + threadIdx.x;
    const int node = gid >> 5;
    const int lane = gid & 31;
    if (node >= n_nodes) return;

    const float* a = acc + (size_t)node * HIDDEN;
    float sum = a[lane] * Wout[lane] + a[lane + 32] * Wout[lane + 32];
    #pragma unroll
    for (int off = 16; off > 0; off >>= 1)
        sum += __shfl_xor(sum, off, 32);

    if (lane == 0) out[node] = sum + bout[0];
}

// ---------------------------------------------------------------------------
extern "C" void kernel_launch(void* const* d_in, const int* in_sizes, int n_in,
                              void* d_out, int out_size, void* d_ws, size_t ws_size,
                              hipStream_t stream)
{
    const float* x      = (const float*)d_in[0];  // (100000, 4)
    const int*   edges  = (const int*)  d_in[1];  // (1000000, 2)
    const float* W1     = (const float*)d_in[2];  // (4, 64)
    const float* b1     = (const float*)d_in[3];  // (64,)
    const float* phases = (const float*)d_in[4];  // (1000000,)
    const float* Wout   = (const float*)d_in[5];  // (64, 1)
    const float* bout   = (const float*)d_in[6];  // (1,)
    float*       out    = (float*)d_out;          // (100000, 1)

    const size_t hbytes = (size_t)N_NODES * HIDDEN * sizeof(float); // 25.6 MB
    float* h   = (float*)d_ws;
    float* acc = (float*)((char*)d_ws + hbytes);

    // Phase 1: 6250 row-tiles -> 6250 waves -> 200000 threads
    {
        const int waves   = N_NODES / 16;           // 100000 % 16 == 0
        const int threads = waves * 32;
        const int block   = 256;
        const int grid    = (threads + block - 1) / block;
        sheaf_h_wmma<<<grid, block, 0, stream>>>(x, W1, b1, h, acc, N_NODES);
    }

    // Phase 2: one wave per edge
    {
        const long long threads = (long long)N_EDGES * 32;
        const int block = 256;
        const int grid  = (int)((threads + block - 1) / block);
        sheaf_edge_diffuse<<<grid, block, 0, stream>>>(h, edges, phases, acc, N_EDGES);
    }

    // Phase 3: one wave per node
    {
        const long long threads = (long long)N_NODES * 32;
        const int block = 256;
        const int grid  = (int)((threads + block - 1) / block);
        sheaf_project<<<grid, block, 0, stream>>>(acc, Wout, bout, out, N_NODES);
    }
}